// PosEncoding_63221918597563
// MI455X (gfx1250) — compile-verified
//
#include <hip/hip_runtime.h>

// PosEncoding add, fused over the three tensors:
//   out_lang[b,l,:]    = lang[b,l,:]    + pe[l]              * (1/32)
//   out_frames[b,l,:]  = frames[b,l,:]  + pe[lens_lang[b]+l] * (1/32)
//   out_actions[b,l,:] = actions[b,l,:] + pe[lens_lang[b]+l] * (1/32)
// HBM-bound: 768 MB traffic -> ~33 us floor at 23.3 TB/s. One block per (b,l)
// moves 12 KB in / 12 KB out with 3 NT b128 streams in flight per wave and two
// CDNA5 async LDS stages for the (L2-resident) pe rows.

#define D_MODEL   1024
#define SEQ       512
#define BATCH     64
#define BLOCK     256
#define PER_TENSOR ((size_t)BATCH * SEQ * D_MODEL)   // 33,554,432 floats

typedef float v4f __attribute__((ext_vector_type(4)));

__global__ __launch_bounds__(BLOCK) void pe_add_kernel(
    const float* __restrict__ lang,
    const float* __restrict__ frames,
    const float* __restrict__ actions,
    const int*   __restrict__ lens_lang,
    const float* __restrict__ pe,
    float*       __restrict__ out)
{
    __shared__ float lds_enc[2 * D_MODEL];   // 8 KB: row l, then row lens+l

    const unsigned r   = blockIdx.x;         // [0, 64*512)
    const unsigned b   = r >> 9;             // / 512
    const unsigned l   = r & 511u;
    const unsigned tid = threadIdx.x;

    // within-tensor flat offset of this lane's float4
    const size_t off = (size_t)r * D_MODEL + tid * 4u;

    // 1) Issue all three long-latency HBM streaming loads first (non-temporal:
    //    read-once data, keeps L2 clean for the reused pe rows).
    v4f xl = __builtin_nontemporal_load((const v4f*)(lang    + off));
    v4f xf = __builtin_nontemporal_load((const v4f*)(frames  + off));
    v4f xa = __builtin_nontemporal_load((const v4f*)(actions + off));

    // 2) CDNA5 async copies: stage the two 4 KB enc rows into LDS.
    //    frames and actions share row lens_lang[b]+l, so only two stages.
    //    Each lane copies and later reads back its own 16 B slot -> only
    //    s_wait_asynccnt needed, no barrier.
    const unsigned row1 = (unsigned)lens_lang[b] + l;   // block-uniform
    {
        const unsigned lds0 = (unsigned)(size_t)(&lds_enc[0]) + tid * 16u;
        const unsigned lds1 = lds0 + (unsigned)(D_MODEL * sizeof(float));
        const float* g0 = pe + (size_t)l    * D_MODEL + tid * 4u;
        const float* g1 = pe + (size_t)row1 * D_MODEL + tid * 4u;
        asm volatile("global_load_async_to_lds_b128 %0, %1, off"
                     :: "v"(lds0), "v"(g0) : "memory");
        asm volatile("global_load_async_to_lds_b128 %0, %1, off"
                     :: "v"(lds1), "v"(g1) : "memory");
        asm volatile("s_wait_asynccnt 0" ::: "memory");
    }

    v4f e0 = *(const v4f*)(&lds_enc[tid * 4u]);            // enc row l
    v4f e1 = *(const v4f*)(&lds_enc[D_MODEL + tid * 4u]);  // enc row lens+l

    // out = x + e * (1/32)   (1/sqrt(1024) == 0.03125 exactly)
    __builtin_nontemporal_store(xl + e0 * 0.03125f, (v4f*)(out + off));
    __builtin_nontemporal_store(xf + e1 * 0.03125f, (v4f*)(out + PER_TENSOR + off));
    __builtin_nontemporal_store(xa + e1 * 0.03125f, (v4f*)(out + 2 * PER_TENSOR + off));
}

extern "C" void kernel_launch(void* const* d_in, const int* in_sizes, int n_in,
                              void* d_out, int out_size, void* d_ws, size_t ws_size,
                              hipStream_t stream) {
    const float* lang      = (const float*)d_in[0];
    const float* frames    = (const float*)d_in[1];
    const float* actions   = (const float*)d_in[2];
    const int*   lens_lang = (const int*)  d_in[3];
    // d_in[4] = lens_frames : unused by the reference computation
    const float* pe        = (const float*)d_in[5];
    float*       out       = (float*)d_out;

    const unsigned nblocks = BATCH * SEQ;    // 32,768 blocks of 8 wave32s
    pe_add_kernel<<<nblocks, BLOCK, 0, stream>>>(lang, frames, actions,
                                                 lens_lang, pe, out);
}